// VqNet_17660905521352
// MI455X (gfx1250) — compile-verified
//
#include <hip/hip_runtime.h>
#include <hip/hip_bf16.h>
#include <math.h>

typedef __attribute__((ext_vector_type(2))) float v2f;
typedef __attribute__((ext_vector_type(8))) float v8f;

#define KCLS 32

// ---------------- zero the accumulators (base[I] and M[I*K] are contiguous) --
__global__ void vq_zero_f32(float* __restrict__ p, int n) {
    int idx = blockIdx.x * blockDim.x + threadIdx.x;
    if (idx < n) p[idx] = 0.0f;
}

// ---------------- per-worker scalar tables ----------------------------------
// o_j = log((1-s)/K), delta_j = log(s + (1-s)/K) - o_j,  s = sigmoid(l_j)
__global__ void vq_worker_tables(const float* __restrict__ snr,
                                 float* __restrict__ w_o,
                                 float* __restrict__ w_d, int J) {
    int j = blockIdx.x * blockDim.x + threadIdx.x;
    if (j < J) {
        float l = snr[j];
        float s = 1.0f / (1.0f + expf(-l));
        float off = (1.0f - s) * (1.0f / (float)KCLS);
        float o = logf(off);
        float d = logf(s + off);
        w_o[j] = o;
        w_d[j] = d - o;
    }
}

// ---------------- scatter: 2 fp32 atomics per observation into L2 -----------
__global__ void vq_scatter(const int* __restrict__ ii, const int* __restrict__ jj,
                           const int* __restrict__ y,
                           const float* __restrict__ w_o, const float* __restrict__ w_d,
                           float* __restrict__ base, float* __restrict__ Mbuf, int N) {
    int n = blockIdx.x * blockDim.x + threadIdx.x;
    if (n < N) {
        int i = ii[n];
        int j = jj[n];
        int c = y[n];
        atomicAdd(&base[i], w_o[j]);
        atomicAdd(&Mbuf[(size_t)i * KCLS + c], w_d[j]);
    }
}

// ---------------- finalize: softmax + logsumexp, rowsum via WMMA ------------
// One wave32 handles a 16x32 tile (16 tasks). Register layout matches the
// 32-bit 16x4 A-fragment: lane L holds row r=L%16, column pairs 4c+2h,4c+2h+1
// with h=L/16. Rowsum = E(16x32) x Ones(32x16) as 8 chained
// v_wmma_f32_16x16x4_f32 accumulations (pure f32 FMA -> exact f32 math).
__global__ void __launch_bounds__(256)
vq_finalize(const float* __restrict__ base, const float* __restrict__ Mbuf,
            float* __restrict__ qz, float* __restrict__ Vq, int I) {
    const int lane = threadIdx.x & 31;
    const int wave = threadIdx.x >> 5;
    const int r = lane & 15;   // row within tile
    const int h = lane >> 4;   // column-pair selector (K%4 in {2h,2h+1})

    int tile = blockIdx.x * 8 + wave;
    int grow = tile * 16 + r;
    bool valid = (grow < I);
    int row = valid ? grow : (I - 1);   // clamp loads; keep EXEC uniform

    float b = base[row];
    const float* mrow = Mbuf + (size_t)row * KCLS;

    float m[16];
    float mx = -INFINITY;
#pragma unroll
    for (int c4 = 0; c4 < 8; ++c4) {
        int col = 4 * c4 + 2 * h;                      // even -> 8B aligned
        v2f mv = *(const v2f*)(mrow + col);
        float m0 = b + mv.x;
        float m1 = b + mv.y;
        m[2 * c4]     = m0;
        m[2 * c4 + 1] = m1;
        mx = fmaxf(mx, fmaxf(m0, m1));
    }
    // full-row max: lanes L and L^16 share a row with complementary columns
    mx = fmaxf(mx, __shfl_xor(mx, 16, 32));

    float e[16];
#pragma unroll
    for (int k = 0; k < 16; ++k) e[k] = expf(m[k] - mx);

    // rowsum via 8x V_WMMA_F32_16X16X4_F32 with B = ones (all columns of D
    // end up equal to the row sum)
    v8f acc = {};
    const v2f ones = {1.0f, 1.0f};
#pragma unroll
    for (int c4 = 0; c4 < 8; ++c4) {
        v2f a = {e[2 * c4], e[2 * c4 + 1]};
        acc = __builtin_amdgcn_wmma_f32_16x16x4_f32(
            /*neg_a=*/false, a, /*neg_b=*/false, ones,
            /*c_mod=*/(short)0, acc, /*reuse_a=*/false, /*reuse_b=*/false);
    }

    // D layout: lanes 0-15 VGPR v = D[M=v, N=lane]; lanes 16-31 VGPR v = D[M=v+8].
    // Lane 0 publishes S[0..7], lane 16 publishes S[8..15].
    __shared__ float Ssum[8][16];
    if (r == 0) {
#pragma unroll
        for (int v = 0; v < 8; ++v) Ssum[wave][h * 8 + v] = acc[v];
    }
    __syncthreads();
    float S = Ssum[wave][r];
    float inv = 1.0f / S;

    if (valid) {
        float* qrow = qz + (size_t)row * KCLS;
#pragma unroll
        for (int c4 = 0; c4 < 8; ++c4) {
            int col = 4 * c4 + 2 * h;
            v2f qv = {e[2 * c4] * inv, e[2 * c4 + 1] * inv};
            *(v2f*)(qrow + col) = qv;
        }
        if (h == 0) Vq[row] = mx + logf(S);   // Vq = logsumexp(cll) exactly
    }
}

extern "C" void kernel_launch(void* const* d_in, const int* in_sizes, int n_in,
                              void* d_out, int out_size, void* d_ws, size_t ws_size,
                              hipStream_t stream) {
    const float* snr = (const float*)d_in[0];
    const int*   ii  = (const int*)d_in[1];
    const int*   jj  = (const int*)d_in[2];
    const int*   y   = (const int*)d_in[3];

    const int J = in_sizes[0];
    const int N = in_sizes[1];
    const int I = out_size / (KCLS + 1);   // out = qz (I*K) ++ Vq (I)

    // workspace layout: [w_o J][w_d J][base I][M I*K]  (base & M contiguous)
    float* ws   = (float*)d_ws;
    float* w_o  = ws;
    float* w_d  = ws + J;
    float* base = ws + 2 * (size_t)J;
    float* Mbuf = base + I;

    float* qz = (float*)d_out;
    float* Vq = qz + (size_t)I * KCLS;

    int zn = I * (KCLS + 1);
    vq_zero_f32<<<(zn + 255) / 256, 256, 0, stream>>>(base, zn);
    vq_worker_tables<<<(J + 255) / 256, 256, 0, stream>>>(snr, w_o, w_d, J);
    vq_scatter<<<(N + 255) / 256, 256, 0, stream>>>(ii, jj, y, w_o, w_d, base, Mbuf, N);

    int nTiles = (I + 15) / 16;
    vq_finalize<<<(nTiles + 7) / 8, 256, 0, stream>>>(base, Mbuf, qz, Vq, I);
}